// GAT_Classification_Net_46394236731697
// MI455X (gfx1250) — compile-verified
//
#include <hip/hip_runtime.h>
#include <cstdint>

typedef __attribute__((ext_vector_type(2))) float v2f;
typedef __attribute__((ext_vector_type(8))) float v8f;

// Global-address-space pointer: guarantees global_load_* (not flat_load_*)
typedef const float __attribute__((address_space(1)))* gcfp;

#define NEG_SLOPE 0.2f

// ---------------------------------------------------------------------------
// GEMM: H[M,Nout] = maybe_relu(A[M,K]) @ W[K,Nout] using V_WMMA_F32_16X16X4_F32
// One 16x64 output tile per wave32 (4 accumulators), K stepped by 4.
// A fragment per lane: row = lane&15, cols (k+2h, k+2h+1), h = lane>>4 -> b64 load
// B fragment per lane: col = lane&15 (+16*t), rows (k+2h, k+2h+1)
// C layout: VGPR r -> row r (lanes 0-15) / row r+8 (lanes 16-31), col = lane&15
// B addressing: 8 per-lane AS(1) pointers bumped by 4*Nout per step; all 8 loads
// issued before the 4 WMMAs so they clause together and overlap the matrix ops.
// ---------------------------------------------------------------------------
template <bool RELU>
__global__ void gemm_wmma_f32(const float* __restrict__ A, const float* __restrict__ W,
                              float* __restrict__ H, int M, int K, int Nout,
                              int tilesN64, int totalTiles)
{
    int wavesPerBlock = blockDim.x >> 5;
    int wid  = blockIdx.x * wavesPerBlock + (threadIdx.x >> 5);
    if (wid >= totalTiles) return;          // wave-uniform exit: EXEC stays full
    int lane = threadIdx.x & 31;
    int hl   = lane >> 4;                   // half-wave select
    int l    = lane & 15;
    int tm = wid / tilesN64, tn = wid % tilesN64;
    int m0 = tm * 16, n0 = tn * 64;

    int  colB[4];
    bool colOK[4];
    gcfp bp[8];                             // [2t]: row k+2hl, [2t+1]: row k+2hl+1
    #pragma unroll
    for (int t = 0; t < 4; ++t) {
        colB[t]  = n0 + 16 * t + l;
        colOK[t] = colB[t] < Nout;
        int colC = colOK[t] ? colB[t] : (Nout - 1); // clamp: no EXEC divergence
        bp[2 * t]     = (gcfp)(W + (size_t)(2 * hl) * Nout + colC);
        bp[2 * t + 1] = bp[2 * t] + Nout;
    }
    const size_t bstep = (size_t)4 * Nout;

    gcfp ap = (gcfp)(A + (size_t)(m0 + l) * K + 2 * hl); // 8B-aligned (K%4==0)
    v8f c[4];
    #pragma unroll
    for (int t = 0; t < 4; ++t) c[t] = (v8f){0.f,0.f,0.f,0.f,0.f,0.f,0.f,0.f};

    for (int k = 0; k < K; k += 4) {
        v2f av = { ap[0], ap[1] };          // global_load_b64
        ap += 4;
        if (RELU) { av.x = fmaxf(av.x, 0.f); av.y = fmaxf(av.y, 0.f); }
        v2f bv[4];
        #pragma unroll
        for (int t = 0; t < 4; ++t)         // 8 global_load_b32, clause-friendly
            bv[t] = (v2f){ *bp[2 * t], *bp[2 * t + 1] };
        #pragma unroll
        for (int t = 0; t < 4; ++t)
            // (neg_a, A, neg_b, B, c_mod, C, reuse_a, reuse_b)
            c[t] = __builtin_amdgcn_wmma_f32_16x16x4_f32(false, av, false, bv[t],
                                                         (short)0, c[t], false, false);
        #pragma unroll
        for (int j = 0; j < 8; ++j) bp[j] += bstep;
    }

    #pragma unroll
    for (int t = 0; t < 4; ++t) {
        if (colOK[t]) {
            #pragma unroll
            for (int r = 0; r < 8; ++r) {
                int row = m0 + r + 8 * hl;
                H[(size_t)row * Nout + colB[t]] = c[t][r];
            }
        }
    }
}

// ---------------------------------------------------------------------------
// Per-node attention coefficients: as[i] = h[i,:].a_src ; ad[i] = h[i,:].a_dst
// ---------------------------------------------------------------------------
__global__ void alpha_kernel(const float* __restrict__ H,
                             const float* __restrict__ aS, const float* __restrict__ aD,
                             float* __restrict__ asOut, float* __restrict__ adOut,
                             int Nn, int D)
{
    int wavesPerBlock = blockDim.x >> 5;
    int wid    = blockIdx.x * wavesPerBlock + (threadIdx.x >> 5);
    int lane   = threadIdx.x & 31;
    int stride = gridDim.x * wavesPerBlock;
    for (int i = wid; i < Nn; i += stride) {
        const float* hr = H + (size_t)i * D;
        float ss = 0.f, sd = 0.f;
        for (int cIdx = lane; cIdx < D; cIdx += 32) {
            float v = hr[cIdx];
            ss += v * aS[cIdx];
            sd += v * aD[cIdx];
        }
        #pragma unroll
        for (int off = 16; off > 0; off >>= 1) {
            ss += __shfl_xor(ss, off, 32);
            sd += __shfl_xor(sd, off, 32);
        }
        if (lane == 0) { asOut[i] = ss; adOut[i] = sd; }
    }
}

__global__ void init_softmax_kernel(float* __restrict__ m, float* __restrict__ den, int Nn)
{
    for (int i = blockIdx.x * blockDim.x + threadIdx.x; i < Nn;
         i += gridDim.x * blockDim.x) {
        m[i]   = -INFINITY;
        den[i] = 0.f;
    }
}

__global__ void init_bias_kernel(float* __restrict__ agg, const float* __restrict__ b,
                                 long long total, int D)
{
    for (long long i = (long long)blockIdx.x * blockDim.x + threadIdx.x; i < total;
         i += (long long)gridDim.x * blockDim.x) {
        agg[i] = b[i % D];
    }
}

__device__ __forceinline__ float leaky(float v) { return v > 0.f ? v : NEG_SLOPE * v; }

__device__ __forceinline__ void atomicMaxF(float* addr, float value)
{
    if (value >= 0.f) atomicMax((int*)addr, __float_as_int(value));
    else              atomicMin((unsigned int*)addr, __float_as_uint(value));
}

__global__ void edge_max_kernel(const float* __restrict__ asN, const float* __restrict__ adN,
                                const int* __restrict__ src, const int* __restrict__ dst,
                                int E, int Nn, float* __restrict__ m)
{
    int tot = E + Nn;
    for (int e = blockIdx.x * blockDim.x + threadIdx.x; e < tot;
         e += gridDim.x * blockDim.x) {
        int s, d;
        if (e < E) { s = src[e]; d = dst[e]; } else { s = e - E; d = s; }
        atomicMaxF(m + d, leaky(asN[s] + adN[d]));
    }
}

__global__ void edge_exp_kernel(const float* __restrict__ asN, const float* __restrict__ adN,
                                const int* __restrict__ src, const int* __restrict__ dst,
                                int E, int Nn, const float* __restrict__ m,
                                float* __restrict__ den, float* __restrict__ exb)
{
    int tot = E + Nn;
    for (int e = blockIdx.x * blockDim.x + threadIdx.x; e < tot;
         e += gridDim.x * blockDim.x) {
        int s, d;
        if (e < E) { s = src[e]; d = dst[e]; } else { s = e - E; d = s; }
        float ex = __expf(leaky(asN[s] + adN[d]) - m[d]);
        exb[e] = ex;
        atomicAdd(den + d, ex);
    }
}

// One wave per edge: agg[dst,:] += (ex/denom[dst]) * h[src,:]
// float4 gather path when D % 128 == 0 (layers 1-2), scalar fallback otherwise.
__global__ void scatter_kernel(const float* __restrict__ H,
                               const int* __restrict__ src, const int* __restrict__ dst,
                               int E, int Nn, const float* __restrict__ exb,
                               const float* __restrict__ den, float* __restrict__ agg, int D)
{
    int wavesPerBlock = blockDim.x >> 5;
    int wid    = blockIdx.x * wavesPerBlock + (threadIdx.x >> 5);
    int lane   = threadIdx.x & 31;
    int stride = gridDim.x * wavesPerBlock;
    int tot = E + Nn;
    bool vec = (D & 127) == 0;
    for (int e = wid; e < tot; e += stride) {
        int s, d;
        if (e < E) { s = src[e]; d = dst[e]; } else { s = e - E; d = s; }
        float coef = exb[e] / den[d];
        const float* hs = H + (size_t)s * D;
        float*       od = agg + (size_t)d * D;
        if (vec) {
            int iters = D >> 7;                        // 128 floats per wave per iter
            for (int i = 0; i < iters; ++i) {
                int base = i * 128 + lane * 4;
                float4 v = *(const float4*)(hs + base); // global_load_b128
                atomicAdd(od + base + 0, coef * v.x);
                atomicAdd(od + base + 1, coef * v.y);
                atomicAdd(od + base + 2, coef * v.z);
                atomicAdd(od + base + 3, coef * v.w);
            }
        } else {
            for (int cIdx = lane; cIdx < D; cIdx += 32)
                atomicAdd(od + cIdx, coef * hs[cIdx]);
        }
    }
}

// One wave per node log_softmax over D columns
__global__ void log_softmax_kernel(const float* __restrict__ agg, float* __restrict__ out,
                                   int Nn, int D)
{
    int wavesPerBlock = blockDim.x >> 5;
    int wid    = blockIdx.x * wavesPerBlock + (threadIdx.x >> 5);
    int lane   = threadIdx.x & 31;
    int stride = gridDim.x * wavesPerBlock;
    for (int i = wid; i < Nn; i += stride) {
        const float* row = agg + (size_t)i * D;
        float mx = -INFINITY;
        for (int cIdx = lane; cIdx < D; cIdx += 32) mx = fmaxf(mx, row[cIdx]);
        #pragma unroll
        for (int off = 16; off > 0; off >>= 1) mx = fmaxf(mx, __shfl_xor(mx, off, 32));
        float s = 0.f;
        for (int cIdx = lane; cIdx < D; cIdx += 32) s += __expf(row[cIdx] - mx);
        #pragma unroll
        for (int off = 16; off > 0; off >>= 1) s += __shfl_xor(s, off, 32);
        float lg = __logf(s);
        for (int cIdx = lane; cIdx < D; cIdx += 32)
            out[(size_t)i * D + cIdx] = row[cIdx] - mx - lg;
    }
}

// ---------------------------------------------------------------------------
extern "C" void kernel_launch(void* const* d_in, const int* in_sizes, int n_in,
                              void* d_out, int out_size, void* d_ws, size_t ws_size,
                              hipStream_t stream)
{
    const int F_IN = 128, H1 = 512, H2 = 256, NC = 40;

    const float* x   = (const float*)d_in[0];
    const int*   ei  = (const int*)d_in[1];
    const float* W1  = (const float*)d_in[2];
    const float* a1s = (const float*)d_in[3];
    const float* a1d = (const float*)d_in[4];
    const float* b1  = (const float*)d_in[5];
    const float* W2  = (const float*)d_in[6];
    const float* a2s = (const float*)d_in[7];
    const float* a2d = (const float*)d_in[8];
    const float* b2  = (const float*)d_in[9];
    const float* W3  = (const float*)d_in[10];
    const float* a3s = (const float*)d_in[11];
    const float* a3d = (const float*)d_in[12];
    const float* b3  = (const float*)d_in[13];

    const int Nn = in_sizes[0] / F_IN;   // 20000
    const int E  = in_sizes[1] / 2;      // 320000
    const int* src = ei;
    const int* dst = ei + E;

    float* ws   = (float*)d_ws;
    float* big1 = ws;                              // Nn * 512
    float* big2 = big1 + (size_t)Nn * H1;          // Nn * 512
    float* asb  = big2 + (size_t)Nn * H1;          // Nn
    float* adb  = asb + Nn;                        // Nn
    float* mb   = adb + Nn;                        // Nn
    float* db   = mb + Nn;                         // Nn
    float* exb  = db + Nn;                         // E + Nn

    auto run_layer = [&](const float* in, int Din, int Dout,
                         const float* W, const float* aS, const float* aD, const float* bias,
                         float* hbuf, float* aggbuf, bool relu) {
        // GEMM (WMMA f32, 16x64 tile per wave)
        int tilesM   = Nn / 16;                     // 20000 % 16 == 0
        int tilesN64 = (Dout + 63) / 64;
        int total    = tilesM * tilesN64;
        int wavesPB  = 4;
        int blocks   = (total + wavesPB - 1) / wavesPB;
        if (relu)
            gemm_wmma_f32<true><<<blocks, 128, 0, stream>>>(in, W, hbuf, Nn, Din, Dout,
                                                            tilesN64, total);
        else
            gemm_wmma_f32<false><<<blocks, 128, 0, stream>>>(in, W, hbuf, Nn, Din, Dout,
                                                             tilesN64, total);
        // attention coefficients
        alpha_kernel<<<1024, 256, 0, stream>>>(hbuf, aS, aD, asb, adb, Nn, Dout);
        // softmax state + bias preload
        init_softmax_kernel<<<(Nn + 255) / 256, 256, 0, stream>>>(mb, db, Nn);
        long long aggTotal = (long long)Nn * Dout;
        init_bias_kernel<<<4096, 256, 0, stream>>>(aggbuf, bias, aggTotal, Dout);
        // edge softmax
        edge_max_kernel<<<2048, 256, 0, stream>>>(asb, adb, src, dst, E, Nn, mb);
        edge_exp_kernel<<<2048, 256, 0, stream>>>(asb, adb, src, dst, E, Nn, mb, db, exb);
        // weighted scatter aggregation
        scatter_kernel<<<8192, 256, 0, stream>>>(hbuf, src, dst, E, Nn, exb, db, aggbuf, Dout);
    };

    // Layer 1: x -> h1(big1) -> agg1(big2)
    run_layer(x, F_IN, H1, W1, a1s, a1d, b1, big1, big2, false);
    // Layer 2: relu(agg1=big2) -> h2(big1 lo) -> agg2(big1 hi)
    float* h2   = big1;
    float* agg2 = big1 + (size_t)Nn * H2;
    run_layer(big2, H1, H2, W2, a2s, a2d, b2, h2, agg2, true);
    // Layer 3: relu(agg2) -> h3(big2 lo) -> agg3(big2 + Nn*64)
    float* h3   = big2;
    float* agg3 = big2 + (size_t)Nn * 64;
    run_layer(agg2, H2, NC, W3, a3s, a3d, b3, h3, agg3, true);

    // log_softmax -> d_out
    log_softmax_kernel<<<2500, 256, 0, stream>>>(agg3, (float*)d_out, Nn, NC);
}